// GraphSAGE_61186104099703
// MI455X (gfx1250) — compile-verified
//
#include <hip/hip_runtime.h>
#include <hip/hip_bf16.h>
#include <stdint.h>

// ---------------- problem constants (match reference) ----------------
#define NNODES 100000
#define NEDGES 1600000
#define DIN    64
#define DH     128
#define DEMB   64
#define NGRAPH 64

typedef __attribute__((ext_vector_type(2))) float v2f;
typedef __attribute__((ext_vector_type(8))) float v8f;

// ---------------------------------------------------------------------
// utility kernels
// ---------------------------------------------------------------------
__global__ void zero_i32(int* __restrict__ p, int n) {
  int i = blockIdx.x * blockDim.x + threadIdx.x;
  if (i < n) p[i] = 0;
}

__global__ void deg_count_k(const int* __restrict__ ei, int* __restrict__ cnt) {
  int e = blockIdx.x * blockDim.x + threadIdx.x;
  if (e < NEDGES) atomicAdd(&cnt[ei[NEDGES + e]], 1);   // dst row of edge_index
}

// ------------------- 3-kernel exclusive scan (TILE=2048) -------------
__global__ __launch_bounds__(256)
void scan_part_k(const int* __restrict__ cnt, int* __restrict__ outv,
                 int* __restrict__ partials, int n) {
  __shared__ int sd[256];
  int tid  = threadIdx.x;
  int base = blockIdx.x * 2048 + tid * 8;
  int v[8]; int s = 0;
#pragma unroll
  for (int i = 0; i < 8; ++i) { int idx = base + i; v[i] = (idx < n) ? cnt[idx] : 0; s += v[i]; }
  sd[tid] = s; __syncthreads();
  for (int off = 1; off < 256; off <<= 1) {
    int t = (tid >= off) ? sd[tid - off] : 0;
    __syncthreads();
    sd[tid] += t;
    __syncthreads();
  }
  int excl = sd[tid] - s;   // exclusive prefix of this thread
#pragma unroll
  for (int i = 0; i < 8; ++i) { int idx = base + i; if (idx < n) outv[idx] = excl; excl += v[i]; }
  if (tid == 255) partials[blockIdx.x] = sd[255];
}

__global__ __launch_bounds__(256)
void scan_tot_k(int* __restrict__ partials, int nb) {
  __shared__ int sd[256];
  int tid = threadIdx.x;
  int v = (tid < nb) ? partials[tid] : 0;
  sd[tid] = v; __syncthreads();
  for (int off = 1; off < 256; off <<= 1) {
    int t = (tid >= off) ? sd[tid - off] : 0;
    __syncthreads();
    sd[tid] += t;
    __syncthreads();
  }
  if (tid < nb) partials[tid] = sd[tid] - v;
}

__global__ __launch_bounds__(256)
void scan_add_k(int* __restrict__ rp, const int* __restrict__ partials,
                const int* __restrict__ cnt, int n) {
  int tid  = threadIdx.x;
  int base = blockIdx.x * 2048 + tid * 8;
  int off  = partials[blockIdx.x];
#pragma unroll
  for (int i = 0; i < 8; ++i) {
    int idx = base + i;
    if (idx < n) {
      rp[idx] += off;
      if (idx == n - 1) rp[n] = rp[idx] + cnt[idx];   // row_ptr[N] = E
    }
  }
}

__global__ void fill_csr_k(const int* __restrict__ ei, const int* __restrict__ rp,
                           int* __restrict__ fill, int* __restrict__ col) {
  int e = blockIdx.x * blockDim.x + threadIdx.x;
  if (e >= NEDGES) return;
  int s = ei[e];
  int d = ei[NEDGES + e];
  int p = atomicAdd(&fill[d], 1);
  col[rp[d] + p] = s;
}

// sort each adjacency list so the gather sum order is deterministic
__global__ void sort_adj_k(const int* __restrict__ rp, int* __restrict__ col) {
  int n = blockIdx.x * blockDim.x + threadIdx.x;
  if (n >= NNODES) return;
  int b = rp[n], e = rp[n + 1];
  for (int i = b + 1; i < e; ++i) {
    int key = col[i]; int j = i - 1;
    while (j >= b && col[j] > key) { col[j + 1] = col[j]; --j; }
    col[j + 1] = key;
  }
}

__global__ void gcount_k(const int* __restrict__ batch, int* __restrict__ gcount) {
  int i = blockIdx.x * blockDim.x + threadIdx.x;
  if (i < NNODES) atomicAdd(&gcount[batch[i]], 1);
}

__global__ void gscan_k(const int* __restrict__ gcount, int* __restrict__ gstart) {
  if (threadIdx.x == 0 && blockIdx.x == 0) {
    int s = 0;
    for (int g = 0; g < NGRAPH; ++g) { gstart[g] = s; s += gcount[g]; }
    gstart[NGRAPH] = s;
  }
}

// ---------------------------------------------------------------------
// fused SAGE layer: CSR mean-gather into LDS + dual f32 WMMA GEMM + relu
//   hout[16x128 tile] = relu( meanTile @ Wl + xTile @ Wr + b )
// one block = 16 nodes; 8 waves, each wave -> one 16x16 WMMA output tile
// ---------------------------------------------------------------------
template <int K>
__global__ __launch_bounds__(256)
void sage_layer_k(const float* __restrict__ hin, const int* __restrict__ rp,
                  const int* __restrict__ col, const float* __restrict__ Wl,
                  const float* __restrict__ Wr, const float* __restrict__ bias,
                  float* __restrict__ hout) {
  constexpr int RS  = K + 4;     // padded LDS row stride (floats) -> no bank conflicts
  constexpr int VPL = K / 32;    // features per lane in the gather (2 or 4)
  __shared__ float s_mean[16 * RS];
  __shared__ float s_x[16 * RS];

  const int m0   = blockIdx.x * 16;
  const int tid  = threadIdx.x;
  const int lane = tid & 31;
  const int wave = tid >> 5;     // 0..7 (wave32)

  // stage x-tile (16 x K) into LDS, coalesced along K
  for (int idx = tid; idx < 16 * K; idx += 256) {
    int r = idx / K, c = idx % K;            // K is a power of two -> shifts
    s_x[r * RS + c] = hin[(m0 + r) * K + c];
  }

  // gather mean-tile: each wave handles rows {wave, wave+8}
  for (int rr = 0; rr < 2; ++rr) {
    int r    = wave + rr * 8;
    int node = m0 + r;
    int beg  = rp[node], end = rp[node + 1];
    float acc[VPL];
#pragma unroll
    for (int i = 0; i < VPL; ++i) acc[i] = 0.0f;
    for (int j = beg; j < end; ++j) {
      const float* sp = hin + col[j] * K + lane * VPL;   // 512B/256B coalesced
#pragma unroll
      for (int i = 0; i < VPL; ++i) acc[i] += sp[i];
    }
    int deg = end - beg; if (deg < 1) deg = 1;
    float inv = 1.0f / (float)deg;
#pragma unroll
    for (int i = 0; i < VPL; ++i) s_mean[r * RS + lane * VPL + i] = acc[i] * inv;
  }
  __syncthreads();

  // per-wave 16x16 WMMA tile: D = meanTile@Wl + xTile@Wr   (full f32 precision)
  const int  n0 = wave * 16;
  const bool hi = lane >= 16;
  const int  l  = hi ? lane - 16 : lane;

  v8f accv = {};   // C starts at zero; bias added after

#pragma unroll
  for (int pass = 0; pass < 2; ++pass) {
    const float* A = pass ? s_x : s_mean;
    const float* W = pass ? Wr  : Wl;
    const float* arow = A + l * RS + (hi ? 2 : 0);   // A 16x4 layout: hi lanes hold K=k+2,k+3
    for (int k = 0; k < K; k += 4) {
      v2f a = *(const v2f*)(arow + k);               // 8B-aligned LDS read
      v2f b;
      int kb = k + (hi ? 2 : 0);                     // B 4x16 layout mirrors A
      b.x = W[kb * DH + n0 + l];
      b.y = W[(kb + 1) * DH + n0 + l];
      // D = A(16x4,f32) x B(4x16,f32) + C  -> v_wmma_f32_16x16x4_f32
      accv = __builtin_amdgcn_wmma_f32_16x16x4_f32(
          /*neg_a=*/false, a, /*neg_b=*/false, b,
          /*c_mod=*/(short)0, accv, /*reuse_a=*/false, /*reuse_b=*/false);
    }
  }

  // bias + relu + store (D layout: lo lanes rows 0..7, hi lanes rows 8..15)
  float bv = bias[n0 + l];
#pragma unroll
  for (int j = 0; j < 8; ++j) {
    float v = accv[j] + bv;
    v = v > 0.0f ? v : 0.0f;
    int row = hi ? (8 + j) : j;
    hout[(m0 + row) * DH + n0 + l] = v;
  }
}

// ---------------------------------------------------------------------
// global mean pool per graph (batch is sorted -> contiguous node ranges)
// ---------------------------------------------------------------------
__global__ __launch_bounds__(128)
void pool_k(const float* __restrict__ h, const int* __restrict__ gstart,
            float* __restrict__ pooled) {
  int g = blockIdx.x, f = threadIdx.x;        // f in [0,128)
  int b = gstart[g], e = gstart[g + 1];
  float s = 0.0f;
  for (int n = b; n < e; ++n) s += h[n * DH + f];
  int c = e - b; if (c < 1) c = 1;
  pooled[g * DH + f] = s / (float)c;
}

__global__ void mlp1_k(const float* __restrict__ p, const float* __restrict__ W,
                       const float* __restrict__ b, float* __restrict__ o) {
  int t = blockIdx.x * blockDim.x + threadIdx.x;
  if (t >= NGRAPH * (DH / 2)) return;
  int g = t >> 6, j = t & 63;
  float s = b[j];
  for (int k = 0; k < DH; ++k) s += p[g * DH + k] * W[k * (DH / 2) + j];
  o[t] = s > 0.0f ? s : 0.0f;
}

__global__ void mlp2_k(const float* __restrict__ t1, const float* __restrict__ W,
                       const float* __restrict__ b, float* __restrict__ o) {
  int t = blockIdx.x * blockDim.x + threadIdx.x;
  if (t >= NGRAPH * DEMB) return;
  int g = t >> 6, e = t & 63;
  float s = b[e];
  for (int k = 0; k < (DH / 2); ++k) s += t1[g * (DH / 2) + k] * W[k * DEMB + e];
  o[t] = s;
}

// ---------------------------------------------------------------------
extern "C" void kernel_launch(void* const* d_in, const int* in_sizes, int n_in,
                              void* d_out, int out_size, void* d_ws, size_t ws_size,
                              hipStream_t stream) {
  const float* x      = (const float*)d_in[0];
  const int*   ei     = (const int*)d_in[1];     // [2, E]
  const int*   batch  = (const int*)d_in[2];
  const float* Wl1    = (const float*)d_in[3];
  const float* Wr1    = (const float*)d_in[4];
  const float* b1     = (const float*)d_in[5];
  const float* Wl2    = (const float*)d_in[6];
  const float* Wr2    = (const float*)d_in[7];
  const float* b2     = (const float*)d_in[8];
  const float* Wl3    = (const float*)d_in[9];
  const float* Wr3    = (const float*)d_in[10];
  const float* b3     = (const float*)d_in[11];
  const float* Wlin1  = (const float*)d_in[12];
  const float* blin1  = (const float*)d_in[13];
  const float* Wlin2  = (const float*)d_in[14];
  const float* blin2  = (const float*)d_in[15];
  float* out = (float*)d_out;

  // carve workspace
  char* ws = (char*)d_ws;
  size_t off = 0;
  auto carve = [&](size_t bytes) -> void* {
    void* p = ws + off;
    off = (off + bytes + 255) & ~(size_t)255;
    return p;
  };
  int*   rp       = (int*)carve((NNODES + 1) * sizeof(int));
  int*   cnt      = (int*)carve(NNODES * sizeof(int));
  int*   fill     = (int*)carve(NNODES * sizeof(int));
  int*   partials = (int*)carve(1024 * sizeof(int));
  int*   gcount   = (int*)carve(256 * sizeof(int));
  int*   gstart   = (int*)carve(256 * sizeof(int));
  int*   col      = (int*)carve((size_t)NEDGES * sizeof(int));
  float* bufA     = (float*)carve((size_t)NNODES * DH * sizeof(float));
  float* bufB     = (float*)carve((size_t)NNODES * DH * sizeof(float));
  float* pooled   = (float*)carve((size_t)NGRAPH * DH * sizeof(float));
  float* t1       = (float*)carve((size_t)NGRAPH * (DH / 2) * sizeof(float));
  (void)ws_size; (void)n_in; (void)in_sizes; (void)out_size;

  const int EB = (NEDGES + 255) / 256;     // 6250
  const int NB = (NNODES + 255) / 256;     // 391
  const int SB = (NNODES + 2047) / 2048;   // 49 scan tiles
  const int MB = NNODES / 16;              // 6250 (N is a multiple of 16)

  // ---- build CSR (deterministic: adjacency sorted afterwards) ----
  zero_i32<<<NB, 256, 0, stream>>>(cnt, NNODES);
  zero_i32<<<NB, 256, 0, stream>>>(fill, NNODES);
  zero_i32<<<1, 256, 0, stream>>>(gcount, 256);

  deg_count_k<<<EB, 256, 0, stream>>>(ei, cnt);
  scan_part_k<<<SB, 256, 0, stream>>>(cnt, rp, partials, NNODES);
  scan_tot_k<<<1, 256, 0, stream>>>(partials, SB);
  scan_add_k<<<SB, 256, 0, stream>>>(rp, partials, cnt, NNODES);
  fill_csr_k<<<EB, 256, 0, stream>>>(ei, rp, fill, col);
  sort_adj_k<<<NB, 256, 0, stream>>>(rp, col);

  // ---- graph boundaries for pooling ----
  gcount_k<<<NB, 256, 0, stream>>>(batch, gcount);
  gscan_k<<<1, 32, 0, stream>>>(gcount, gstart);

  // ---- 3 fused SAGE layers (gather + WMMA f32 GEMM + relu) ----
  sage_layer_k<DIN><<<MB, 256, 0, stream>>>(x,    rp, col, Wl1, Wr1, b1, bufA);
  sage_layer_k<DH> <<<MB, 256, 0, stream>>>(bufA, rp, col, Wl2, Wr2, b2, bufB);
  sage_layer_k<DH> <<<MB, 256, 0, stream>>>(bufB, rp, col, Wl3, Wr3, b3, bufA);

  // ---- pool + MLP head ----
  pool_k<<<NGRAPH, 128, 0, stream>>>(bufA, gstart, pooled);
  mlp1_k<<<16, 256, 0, stream>>>(pooled, Wlin1, blin1, t1);
  mlp2_k<<<16, 256, 0, stream>>>(t1, Wlin2, blin2, out);
}